// Or_56281251447398
// MI455X (gfx1250) — compile-verified
//
#include <hip/hip_runtime.h>
#include <stdint.h>

#define BATCH 128
#define IN_F  1024
#define OUT_F 1024

#define BM 32
#define BN 64
#define BK 32
#define NCHUNK (IN_F / BK)   // 32
#define NTHREADS 256

// ---------- CDNA5 async global->LDS copy + ASYNCcnt wait helpers ----------
#if defined(__HIP_DEVICE_COMPILE__)

typedef int v4i_t __attribute__((ext_vector_type(4)));
typedef __attribute__((address_space(1))) v4i_t gbl_v4i;
typedef __attribute__((address_space(3))) v4i_t lds_v4i;

#if __has_builtin(__builtin_amdgcn_s_wait_asynccnt)
#define WAIT_ASYNCCNT(n) __builtin_amdgcn_s_wait_asynccnt(n)
#else
#define WAIT_ASYNCCNT(n) asm volatile("s_wait_asynccnt %0" :: "n"(n) : "memory")
#endif

__device__ __forceinline__ void async_copy_b128(const float* gsrc, float* lds_dst) {
#if __has_builtin(__builtin_amdgcn_global_load_async_to_lds_b128)
  __builtin_amdgcn_global_load_async_to_lds_b128(
      (gbl_v4i*)gsrc,
      (lds_v4i*)lds_dst,
      /*imm offset=*/0, /*cpol=*/0);
#else
  unsigned lds_off =
      (unsigned)(uintptr_t)(__attribute__((address_space(3))) void*)lds_dst;
  asm volatile("global_load_async_to_lds_b128 %0, %1, off"
               :: "v"(lds_off), "v"(gsrc)
               : "memory");
#endif
}

#else  // host parse-only stubs
#define WAIT_ASYNCCNT(n) ((void)0)
__device__ __forceinline__ void async_copy_b128(const float*, float*) {}
#endif

__device__ __forceinline__ float clamp01(float x) {
  return fminf(fmaxf(x, 0.0f), 1.0f);
}

// out[b,o] = max_i min(m[b,i], clip(w[i,o],0,1))
__global__ __launch_bounds__(NTHREADS)
void tropical_maxmin_kernel(const float* __restrict__ M,
                            const float* __restrict__ W,
                            float* __restrict__ Out) {
  __shared__ __align__(16) float mA[2][BM * BK];  // [row][k]   4 KB each
  __shared__ __align__(16) float wB[2][BK * BN];  // [k][col]   8 KB each

  const int tid  = threadIdx.x;
  const int tcol = tid & 15;   // column group: cols tcol*4 .. +3
  const int trow = tid >> 4;   // row group:    rows trow*2 .. +1

  const int rowBase = blockIdx.y * BM;
  const int colBase = blockIdx.x * BN;

  // --- per-thread async-copy assignments (16B granules) ---
  // m tile: 32x32 f32 = 256 float4 segments -> 1 per thread
  const int ms_row = tid >> 3;                 // 0..31
  const int ms_seg = tid & 7;                  // 0..7
  const float* mg  = M + (size_t)(rowBase + ms_row) * IN_F + ms_seg * 4;
  const int m_lds  = ms_row * BK + ms_seg * 4;

  // w tile: 32x64 f32 = 512 float4 segments -> 2 per thread
  const int wk0  = tid >> 4;                   // 0..15
  const int wk1  = wk0 + 16;                   // 16..31
  const int wseg = tid & 15;                   // 0..15
  const float* wg0 = W + (size_t)wk0 * OUT_F + colBase + wseg * 4;
  const float* wg1 = W + (size_t)wk1 * OUT_F + colBase + wseg * 4;
  const int w_lds0 = wk0 * BN + wseg * 4;
  const int w_lds1 = wk1 * BN + wseg * 4;

  const float NEG = -__builtin_inff();
  float4 acc0 = make_float4(NEG, NEG, NEG, NEG);
  float4 acc1 = acc0;

  // prologue: chunk 0 -> buffer 0 (3 async instructions per thread)
  async_copy_b128(mg,  &mA[0][m_lds]);
  async_copy_b128(wg0, &wB[0][w_lds0]);
  async_copy_b128(wg1, &wB[0][w_lds1]);

  for (int c = 0; c < NCHUNK; ++c) {
    const int cur = c & 1;

    if (c + 1 < NCHUNK) {
      const int nxt = cur ^ 1;
      const size_t mo   = (size_t)(c + 1) * BK;
      const size_t woff = (size_t)(c + 1) * BK * OUT_F;
      async_copy_b128(mg + mo,    &mA[nxt][m_lds]);
      async_copy_b128(wg0 + woff, &wB[nxt][w_lds0]);
      async_copy_b128(wg1 + woff, &wB[nxt][w_lds1]);
      WAIT_ASYNCCNT(3);   // chunk c done; chunk c+1 still in flight
    } else {
      WAIT_ASYNCCNT(0);
    }
    __syncthreads();

    // clamp the weight tile once in LDS (hoists clip() out of inner loop)
    {
      float4* wp = reinterpret_cast<float4*>(&wB[cur][tid * 8]);
      float4 a = wp[0], b = wp[1];
      a.x = clamp01(a.x); a.y = clamp01(a.y); a.z = clamp01(a.z); a.w = clamp01(a.w);
      b.x = clamp01(b.x); b.y = clamp01(b.y); b.z = clamp01(b.z); b.w = clamp01(b.w);
      wp[0] = a; wp[1] = b;
    }
    __syncthreads();

    const float* mrow0 = &mA[cur][(trow * 2 + 0) * BK];
    const float* mrow1 = &mA[cur][(trow * 2 + 1) * BK];
    const float* wcol  = &wB[cur][tcol * 4];

#pragma unroll 8
    for (int kk = 0; kk < BK; ++kk) {
      const float4 w4 = *reinterpret_cast<const float4*>(wcol + (size_t)kk * BN);
      const float m0 = mrow0[kk];
      const float m1 = mrow1[kk];
      acc0.x = fmaxf(acc0.x, fminf(m0, w4.x));
      acc0.y = fmaxf(acc0.y, fminf(m0, w4.y));
      acc0.z = fmaxf(acc0.z, fminf(m0, w4.z));
      acc0.w = fmaxf(acc0.w, fminf(m0, w4.w));
      acc1.x = fmaxf(acc1.x, fminf(m1, w4.x));
      acc1.y = fmaxf(acc1.y, fminf(m1, w4.y));
      acc1.z = fmaxf(acc1.z, fminf(m1, w4.z));
      acc1.w = fmaxf(acc1.w, fminf(m1, w4.w));
    }
    __syncthreads();
  }

  const int r0 = rowBase + trow * 2;
  const int c0 = colBase + tcol * 4;
  *reinterpret_cast<float4*>(Out + (size_t)r0 * OUT_F + c0)       = acc0;
  *reinterpret_cast<float4*>(Out + (size_t)(r0 + 1) * OUT_F + c0) = acc1;
}

extern "C" void kernel_launch(void* const* d_in, const int* in_sizes, int n_in,
                              void* d_out, int out_size, void* d_ws, size_t ws_size,
                              hipStream_t stream) {
  (void)in_sizes; (void)n_in; (void)out_size; (void)d_ws; (void)ws_size;
  const float* m = (const float*)d_in[0];       // [128, 1024]
  const float* w = (const float*)d_in[1];       // [1024, 1024]
  float* out     = (float*)d_out;               // [128, 1024]

  dim3 grid(OUT_F / BN, BATCH / BM, 1);         // 16 x 4 = 64 blocks
  dim3 block(NTHREADS, 1, 1);
  hipLaunchKernelGGL(tropical_maxmin_kernel, grid, block, 0, stream, m, w, out);
}